// GCNHead_55748675502409
// MI455X (gfx1250) — compile-verified
//
#include <hip/hip_runtime.h>

typedef __attribute__((ext_vector_type(2))) float v2f;
typedef __attribute__((ext_vector_type(8))) float v8f;

#define NN 100000
#define NE 3200000
#define DD 256
#define NL 3
#define BN_EPS 1e-5f

// ---------------- utility kernels ----------------

__global__ void zero_f32(float* __restrict__ p, long n) {
    long i = (long)blockIdx.x * blockDim.x + threadIdx.x;
    if (i < n) p[i] = 0.0f;
}

__global__ void deg_accum(const int* __restrict__ src, const int* __restrict__ dst,
                          float* __restrict__ dout, float* __restrict__ din) {
    int e = blockIdx.x * blockDim.x + threadIdx.x;
    if (e < NE) {
        atomicAdd(&dout[src[e]], 1.0f);
        atomicAdd(&din[dst[e]], 1.0f);
    }
}

__global__ void deg_finalize(float* __restrict__ dout, float* __restrict__ din) {
    int i = blockIdx.x * blockDim.x + threadIdx.x;
    if (i < NN) {
        dout[i] = rsqrtf(fmaxf(dout[i], 1.0f));   // becomes inv_sqrt_out
        din[i]  = rsqrtf(fmaxf(din[i], 1.0f));    // becomes inv_sqrt_in
    }
}

// hs = h * inv_sqrt_out[row], vectorized float4
__global__ void scale_rows(const float* __restrict__ h, const float* __restrict__ inv_out,
                           float* __restrict__ hs) {
    long i = (long)blockIdx.x * blockDim.x + threadIdx.x;   // over NN*DD/4
    if (i < (long)NN * DD / 4) {
        int n = (int)(i >> 6);                   // (i*4)/256
        float s = inv_out[n];
        float4 v = ((const float4*)h)[i];
        v.x *= s; v.y *= s; v.z *= s; v.w *= s;
        ((float4*)hs)[i] = v;
    }
}

// agg[dst[e]] += hs[src[e]]  (edge-parallel, 4 features per thread)
__global__ void edge_scatter(const float* __restrict__ hs, const int* __restrict__ src,
                             const int* __restrict__ dst, float* __restrict__ agg) {
    long t = (long)blockIdx.x * blockDim.x + threadIdx.x;   // NE * 64 threads
    if (t >= (long)NE * (DD / 4)) return;
    int e = (int)(t >> 6);
    int c = (int)((t & 63) << 2);
    int s = src[e], d = dst[e];
    float4 v = *(const float4*)&hs[(long)s * DD + c];
    float* ap = &agg[(long)d * DD + c];
    atomicAdd(ap + 0, v.x);
    atomicAdd(ap + 1, v.y);
    atomicAdd(ap + 2, v.z);
    atomicAdd(ap + 3, v.w);
}

// out = (agg * inv_in[row]) @ W + h @ Wl + h + b + bl
// one block = 16 output rows; 4 waves = 4 x (16x64) column slabs.
// A tiles (16x256 of agg and of h, 16KB each = contiguous rows) are staged
// into LDS once per block via async global->LDS copies (ASYNCcnt), then the
// WMMA A fragments come from LDS (ds_load_b64).
__global__ __launch_bounds__(128)
void gemm_fused(const float* __restrict__ agg, const float* __restrict__ inv_in,
                const float* __restrict__ h,
                const float* __restrict__ W,  const float* __restrict__ Wl,
                const float* __restrict__ b,  const float* __restrict__ bl,
                float* __restrict__ out) {
    __shared__ float sAgg[16 * DD];
    __shared__ float sH[16 * DD];

    const int tid  = threadIdx.x;
    const int lane = tid & 31;
    const int wave = tid >> 5;
    const int row0 = blockIdx.x * 16;          // 6250 * 16 = 100000 exactly
    const int col0 = wave * 64;
    const int m    = lane & 15;                // A row / B-C column within tile
    const int kOff = (lane >> 4) * 2;          // lanes 0-15 -> K 0,1 ; 16-31 -> K 2,3

    // ---- async stage: 16 contiguous rows (16KB) per matrix into LDS ----
    const float* gA = agg + (size_t)row0 * DD;
    const float* gH = h   + (size_t)row0 * DD;
    #pragma unroll
    for (int i = 0; i < 8; ++i) {
        const int f4 = tid + i * 128;                       // float4 index in tile
        const unsigned gOff = (unsigned)(f4 * 16);          // global byte offset
        const unsigned l0 = (unsigned)(unsigned long long)(const void*)&sAgg[f4 * 4];
        const unsigned l1 = (unsigned)(unsigned long long)(const void*)&sH[f4 * 4];
        asm volatile("global_load_async_to_lds_b128 %0, %1, %2"
                     :: "v"(l0), "v"(gOff), "s"(gA) : "memory");
        asm volatile("global_load_async_to_lds_b128 %0, %1, %2"
                     :: "v"(l1), "v"(gOff), "s"(gH) : "memory");
    }
    asm volatile("s_wait_asynccnt 0" ::: "memory");
    __syncthreads();

    const float sIn = inv_in[row0 + m];

    v8f acc[4];
    #pragma unroll
    for (int j = 0; j < 4; ++j) acc[j] = (v8f){};

    for (int k0 = 0; k0 < DD; k0 += 4) {
        // A fragments (16x4 f32: v2f per lane) from LDS, reused across 4 N-subtiles
        v2f a1 = *(const v2f*)&sAgg[m * DD + k0 + kOff];
        a1.x *= sIn; a1.y *= sIn;
        v2f a2 = *(const v2f*)&sH[m * DD + k0 + kOff];
        const int kr = (k0 + kOff) * DD;
        #pragma unroll
        for (int j = 0; j < 4; ++j) {
            const int col = col0 + j * 16 + m;
            v2f b1, b2;
            b1.x = W [kr + col];  b1.y = W [kr + DD + col];
            b2.x = Wl[kr + col];  b2.y = Wl[kr + DD + col];
            acc[j] = __builtin_amdgcn_wmma_f32_16x16x4_f32(
                false, a1, false, b1, (short)0, acc[j], false, false);
            acc[j] = __builtin_amdgcn_wmma_f32_16x16x4_f32(
                false, a2, false, b2, (short)0, acc[j], false, false);
        }
    }

    // C/D layout: VGPR r, lanes 0-15 -> M=r ; lanes 16-31 -> M=r+8 ; N = lane&15
    const int rAdd = (lane >> 4) * 8;
    #pragma unroll
    for (int j = 0; j < 4; ++j) {
        const int col = col0 + j * 16 + m;
        const float bias = b[col] + bl[col];
        #pragma unroll
        for (int r = 0; r < 8; ++r) {
            const int row = rAdd + r;
            out[(long)(row0 + row) * DD + col] = acc[j][r] + sH[row * DD + col] + bias;
        }
    }
}

// per-feature sum and sum-of-squares over all rows
__global__ void colsum(const float* __restrict__ out, float* __restrict__ sums,
                       float* __restrict__ sumsq) {
    int f = threadIdx.x;                       // 256 threads = 256 features
    float s = 0.0f, sq = 0.0f;
    for (int n = blockIdx.x; n < NN; n += gridDim.x) {
        float v = out[(long)n * DD + f];
        s += v; sq += v * v;
    }
    atomicAdd(&sums[f], s);
    atomicAdd(&sumsq[f], sq);
}

__global__ void bn_relu(const float* __restrict__ out, const float* __restrict__ sums,
                        const float* __restrict__ sumsq, const float* __restrict__ gamma,
                        const float* __restrict__ beta, float* __restrict__ hout) {
    long i = (long)blockIdx.x * blockDim.x + threadIdx.x;
    if (i >= (long)NN * DD) return;
    int f = (int)(i & (DD - 1));
    float mean = sums[f] * (1.0f / NN);
    float var  = sumsq[f] * (1.0f / NN) - mean * mean;
    float v = gamma[f] * (out[i] - mean) * rsqrtf(var + BN_EPS) + beta[f];
    hout[i] = fmaxf(v, 0.0f);
}

// ---------------- launcher ----------------

extern "C" void kernel_launch(void* const* d_in, const int* in_sizes, int n_in,
                              void* d_out, int out_size, void* d_ws, size_t ws_size,
                              hipStream_t stream) {
    const float* x     = (const float*)d_in[0];
    const float* W     = (const float*)d_in[1];   // [L,D,D]
    const float* b     = (const float*)d_in[2];   // [L,D]
    const float* Wl    = (const float*)d_in[3];   // [L,D,D]
    const float* bl    = (const float*)d_in[4];   // [L,D]
    const float* gamma = (const float*)d_in[5];   // [L,D]
    const float* beta  = (const float*)d_in[6];   // [L,D]
    const int*   src   = (const int*)d_in[7];
    const int*   dst   = (const int*)d_in[8];
    float* hout = (float*)d_out;

    char* ws = (char*)d_ws;
    size_t off = 0;
    auto alloc = [&](size_t bytes) -> void* {
        void* p = ws + off;
        off = (off + bytes + 255) & ~(size_t)255;
        return p;
    };
    const size_t ND = (size_t)NN * DD;
    float* bufA   = (float*)alloc(ND * sizeof(float));   // hs, then pre-BN out
    float* bufB   = (float*)alloc(ND * sizeof(float));   // agg
    float* invOut = (float*)alloc(NN * sizeof(float));
    float* invIn  = (float*)alloc(NN * sizeof(float));
    float* sums   = (float*)alloc(DD * sizeof(float));
    float* sumsq  = (float*)alloc(DD * sizeof(float));

    // degrees -> inv-sqrt (clamped to 1)
    zero_f32<<<(NN + 255) / 256, 256, 0, stream>>>(invOut, NN);
    zero_f32<<<(NN + 255) / 256, 256, 0, stream>>>(invIn, NN);
    deg_accum<<<(NE + 255) / 256, 256, 0, stream>>>(src, dst, invOut, invIn);
    deg_finalize<<<(NN + 255) / 256, 256, 0, stream>>>(invOut, invIn);

    const long nd4 = (long)ND / 4;
    const long et  = (long)NE * (DD / 4);

    const float* h = x;
    for (int l = 0; l < NL; ++l) {
        // hs = h * inv_sqrt_out
        scale_rows<<<(unsigned)((nd4 + 255) / 256), 256, 0, stream>>>(h, invOut, bufA);
        // agg = segment_sum(hs[src] -> dst)
        zero_f32<<<(unsigned)((ND + 255) / 256), 256, 0, stream>>>(bufB, (long)ND);
        edge_scatter<<<(unsigned)((et + 255) / 256), 256, 0, stream>>>(bufA, src, dst, bufB);
        // out = (agg*inv_in) @ W[l] + h @ Wl[l] + h + b[l] + bl[l]   (WMMA + async LDS)
        gemm_fused<<<NN / 16, 128, 0, stream>>>(bufB, invIn, h,
                                                W + (size_t)l * DD * DD,
                                                Wl + (size_t)l * DD * DD,
                                                b + (size_t)l * DD,
                                                bl + (size_t)l * DD, bufA);
        // batch-norm statistics
        zero_f32<<<2, 256, 0, stream>>>(sums, DD);
        zero_f32<<<2, 256, 0, stream>>>(sumsq, DD);
        colsum<<<512, 256, 0, stream>>>(bufA, sums, sumsq);
        // normalize + relu -> h for next layer (in d_out)
        bn_relu<<<(unsigned)((ND + 255) / 256), 256, 0, stream>>>(
            bufA, sums, sumsq, gamma + (size_t)l * DD, beta + (size_t)l * DD, hout);
        h = hout;
    }
}